// DRP_LAYER_41644002902275
// MI455X (gfx1250) — compile-verified
//
#include <hip/hip_runtime.h>

// ---------------------------------------------------------------------------
// FDTD/DRP stepper for MI455X (gfx1250, wave32).
// Memory-bound problem (AI ~2-4 flop/B): f32 end-to-end, coalesced traffic,
// interior stencils executed on the WMMA matrix units as banded matmuls
// (V_WMMA_F32_16X16X4_F32), operands staged via LDS. Boundary strips use
// fully predicated VALU kernels mirroring the reference pad arithmetic.
//
// Round-2 changes: zero-padded LDS coefficient band (kills the divergent
// EXEC-masked ds_loads seen in round-1 asm), load/WMMA phase separation so
// one deep s_wait_dscnt covers a whole 6-WMMA chain, 16B-aligned LDS tiles
// for ds_load_b64 merging, and global_prefetch_b8 in the Faraday kernels.
// ---------------------------------------------------------------------------

constexpr int   N     = 2048;
constexpr int   BATCH = 2;
constexpr float CFL   = 0.35f;

// Interior tiling: 16x16 tiles covering [TILE0, COVER_END]^2 of the E grid.
// There, {central, kf, kb} are uniformly active and {fu, fd} uniformly off,
// so a single 9x4 coefficient band (shared by the y- and x-operators)
// captures the whole interior operator.
constexpr int TILE0     = 5;
constexpr int NTILE     = 127;                     // covers rows/cols 5..2036
constexpr int COVER_END = TILE0 + 16 * NTILE - 1;  // 2036

typedef float v2f __attribute__((ext_vector_type(2)));
typedef float v8f __attribute__((ext_vector_type(8)));

#if defined(__has_builtin)
#if __has_builtin(__builtin_amdgcn_wmma_f32_16x16x4_f32)
#define HAVE_WMMA_F32X4 1
#endif
#endif

// Combined 3x4 DRP kernel  W = YEE + delta*DELTA + beta*BETA + gamma*GAMMA
__device__ __forceinline__ void makeW(float beta, float delta, float gamma, float W[3][4]) {
  const float b = 0.25f * beta, g = 0.1f * gamma;
  W[0][0] = -b + g;  W[0][1] = -0.5f + delta + b - g;  W[0][2] =  b - g;  W[0][3] = -b + g;
  W[1][0] = 0.f;     W[1][1] = -2.f * delta;           W[1][2] = 0.f;     W[1][3] = 0.f;
  W[2][0] =  b - g;  W[2][1] =  0.5f + delta - b + g;  W[2][2] = -b + g;  W[2][3] =  b - g;
}

__device__ __constant__ float d_KF[4]  = {-11.f / 6.f, 3.f, -1.5f, 1.f / 3.f};
__device__ __constant__ float d_KB[4]  = {-1.f / 3.f, 1.5f, -3.f, 11.f / 6.f};
__device__ __constant__ float d_FUV[4] = {-1.f, 3.f, -3.f, 1.f};
__device__ __constant__ float d_FDV[4] = {1.f, -3.f, 3.f, -1.f};
__device__ __constant__ float d_KEF[4] = {-1.5f, 2.f, -0.5f, 0.f};
__device__ __constant__ float d_KEB[4] = {0.f, 0.5f, -2.f, 1.5f};

// ---------------------------------------------------------------------------
// Ampere interior: one wave per 16x16 tile.
//   s2 (y-op on Hx):  Out = Band(16x24) x In(24x16)   -> A = coeffs, B = data
//   s1 (x-op on Hy):  Out = In(16x24) x Band(24x16)   -> A = data,  B = coeffs
// Band rows are stored zero-padded: sCP[d + 15][y], valid band at d in [0,8]
// (central taps at d 3..5, kf folded at y==1 d>=5, kb folded at y==2 d<=3),
// zeros elsewhere, so per-chunk coefficient fetches are unconditional.
// ---------------------------------------------------------------------------
__global__ void __launch_bounds__(32)
amper_tile(const float* __restrict__ E, const float* __restrict__ Hx,
           const float* __restrict__ Hy, float* __restrict__ Eo,
           const float* __restrict__ pb, const float* __restrict__ pd,
           const float* __restrict__ pg) {
  alignas(16) __shared__ float sHx[24][20];  // rows i0-5..i0+18, cols j0-2..j0+17
  alignas(16) __shared__ float sHy[20][24];  // rows i0-2..i0+17, cols j0-5..j0+18
  alignas(16) __shared__ float sCP[40][4];   // zero-padded band, row = d + 15

  const int lane = threadIdx.x;
  const int i0 = TILE0 + 16 * blockIdx.y;
  const int j0 = TILE0 + 16 * blockIdx.x;
  const int b  = blockIdx.z;

  const float* Hxb = Hx + (size_t)b * (N - 1) * N;
  const float* Hyb = Hy + (size_t)b * N * (N - 1);
  const float* Eb  = E  + (size_t)b * (N + 1) * (N + 1);
  float*       Eob = Eo + (size_t)b * (N + 1) * (N + 1);

  float W[3][4];
  makeW(pb[0], pd[0], pg[0], W);

  // zero-padded coefficient band (identical for the transposed x-operator)
  for (int t = lane; t < 40 * 4; t += 32) {
    const int row = t >> 2, y = t & 3;
    const int d = row - 15;
    float v = 0.f;
    if (d >= 3 && d <= 5) v += (d == 3) ? W[0][y] : (d == 4) ? W[1][y] : W[2][y];
    if (y == 1 && d >= 5 && d <= 8) v += d_KF[d - 5];
    if (y == 2 && d >= 0 && d <= 3) v += d_KB[d];
    sCP[row][y] = v;
  }
  for (int t = lane; t < 24 * 20; t += 32) {
    const int r = t / 20, c = t % 20;
    sHx[r][c] = Hxb[(size_t)(i0 - 5 + r) * N + (j0 - 2 + c)];
  }
  for (int t = lane; t < 20 * 24; t += 32) {
    const int r = t / 24, c = t % 24;
    sHy[r][c] = Hyb[(size_t)(i0 - 2 + r) * (N - 1) + (j0 - 5 + c)];
  }
  __syncthreads();

  const int ml = lane & 15;     // A-matrix M / B-matrix N index
  const int hi = lane >> 4;     // K-pair selector: VGPR0/1 hold K = 2*hi, 2*hi+1
  const int kbase = 2 * hi;     // K of first element in this lane's operand pair

#if defined(HAVE_WMMA_F32X4) && defined(__AMDGCN__)
  v8f acc1 = {0.f, 0.f, 0.f, 0.f, 0.f, 0.f, 0.f, 0.f};
  v8f acc2 = {0.f, 0.f, 0.f, 0.f, 0.f, 0.f, 0.f, 0.f};

  // s2: A = band coefficients, B = Hx slab, one pass per x-offset c (0..3)
  #pragma unroll
  for (int c = 0; c < 4; ++c) {
    v2f av[6], bv[6];
    #pragma unroll
    for (int bk = 0; bk < 6; ++bk) {          // load phase: 24 ds_loads pipeline
      const int p0 = 4 * bk + kbase;
      av[bk].x = sCP[p0 - ml + 15][c];
      av[bk].y = sCP[p0 - ml + 16][c];
      bv[bk].x = sHx[p0][ml + c];
      bv[bk].y = sHx[p0 + 1][ml + c];
    }
    #pragma unroll
    for (int bk = 0; bk < 6; ++bk) {          // math phase: back-to-back WMMA
      acc2 = __builtin_amdgcn_wmma_f32_16x16x4_f32(false, av[bk], false, bv[bk],
                                                   (short)0, acc2, false, false);
    }
  }
  // s1: A = Hy slab, B = band coefficients, one pass per y-offset dr (0..3)
  #pragma unroll
  for (int dr = 0; dr < 4; ++dr) {
    v2f av[6], bv[6];
    #pragma unroll
    for (int bk = 0; bk < 6; ++bk) {
      const int p0 = 4 * bk + kbase;
      av[bk].x = sHy[ml + dr][p0];            // adjacent pair -> ds_load_b64
      av[bk].y = sHy[ml + dr][p0 + 1];
      bv[bk].x = sCP[p0 - ml + 15][dr];
      bv[bk].y = sCP[p0 - ml + 16][dr];
    }
    #pragma unroll
    for (int bk = 0; bk < 6; ++bk) {
      acc1 = __builtin_amdgcn_wmma_f32_16x16x4_f32(false, av[bk], false, bv[bk],
                                                   (short)0, acc1, false, false);
    }
  }
  // C/D layout: VGPR v -> M = v (lanes 0-15), M = v+8 (lanes 16-31); N = lane%16
  #pragma unroll
  for (int v = 0; v < 8; ++v) {
    const int m = v + 8 * hi;
    const size_t idx = (size_t)(i0 + m) * (N + 1) + (j0 + ml);
    Eob[idx] = Eb[idx] + CFL * (acc1[v] - acc2[v]);
  }
#else
  // VALU fallback, same tile mapping
  #pragma unroll
  for (int v = 0; v < 8; ++v) {
    const int m = v + 8 * hi, n = ml;
    float s2 = 0.f, s1 = 0.f;
    for (int c = 0; c < 4; ++c)
      for (int di = 0; di < 9; ++di) s2 += sCP[di + 15][c] * sHx[m + di][n + c];
    for (int dr = 0; dr < 4; ++dr)
      for (int ei = 0; ei < 9; ++ei) s1 += sCP[ei + 15][dr] * sHy[m + dr][n + ei];
    const size_t idx = (size_t)(i0 + m) * (N + 1) + (j0 + n);
    Eob[idx] = Eb[idx] + CFL * (s1 - s2);
  }
#endif
}

// ---------------------------------------------------------------------------
// Ampere boundary mop-up: fully predicated per-point form (mirrors every pad)
// ---------------------------------------------------------------------------
__global__ void amper_edge(const float* __restrict__ E, const float* __restrict__ Hx,
                           const float* __restrict__ Hy, float* __restrict__ Eo,
                           const float* __restrict__ pb, const float* __restrict__ pd,
                           const float* __restrict__ pg) {
  const int j = blockIdx.x * blockDim.x + threadIdx.x;
  const int i = blockIdx.y * blockDim.y + threadIdx.y;
  const int b = blockIdx.z;
  if (i > N || j > N) return;
  if (i >= TILE0 && i <= COVER_END && j >= TILE0 && j <= COVER_END) return;

  float W[3][4];
  makeW(pb[0], pd[0], pg[0], W);
  const float* Hxb = Hx + (size_t)b * (N - 1) * N;
  const float* Hyb = Hy + (size_t)b * N * (N - 1);
  const int SY = N - 1;

  float s2 = 0.f, s1 = 0.f;
  // ---- s2 (y-operator on Hx) ----
  if (i >= 2 && i <= N - 2 && j >= 2 && j <= N - 2) {
    #pragma unroll
    for (int r = 0; r < 3; ++r)
      #pragma unroll
      for (int c = 0; c < 4; ++c)
        s2 += W[r][c] * Hxb[(size_t)(i - 2 + r) * N + (j - 2 + c)];
  }
  if (i <= N - 5 && j >= 1)
    #pragma unroll
    for (int r = 0; r < 4; ++r) s2 += d_KF[r] * Hxb[(size_t)(i + r) * N + (j - 1)];
  if (i >= 5 && j <= N - 1)
    #pragma unroll
    for (int r = 0; r < 4; ++r) s2 += d_KB[r] * Hxb[(size_t)(i - 5 + r) * N + j];
  if (i >= 1 && i <= 2 && j >= 2 && j <= N - 2)
    #pragma unroll
    for (int r = 0; r < 4; ++r) s2 += d_FUV[r] * Hxb[(size_t)(i - 1 + r) * N + (j - 1)];
  if (i >= N - 2 && i <= N - 1 && j >= 2 && j <= N - 2)
    #pragma unroll
    for (int r = 0; r < 4; ++r) s2 += d_FDV[r] * Hxb[(size_t)(i - 4 + r) * N + j];
  // ---- s1 (x-operator on Hy, transposed kernels) ----
  if (i >= 2 && i <= N - 2 && j >= 2 && j <= N - 2) {
    #pragma unroll
    for (int r = 0; r < 4; ++r)
      #pragma unroll
      for (int c = 0; c < 3; ++c)
        s1 += W[c][r] * Hyb[(size_t)(i - 2 + r) * SY + (j - 2 + c)];
  }
  if (i >= 1 && j <= N - 5)
    #pragma unroll
    for (int c = 0; c < 4; ++c) s1 += d_KF[c] * Hyb[(size_t)(i - 1) * SY + (j + c)];
  if (i <= N - 1 && j >= 5)
    #pragma unroll
    for (int c = 0; c < 4; ++c) s1 += d_KB[c] * Hyb[(size_t)i * SY + (j - 5 + c)];
  if (i >= 2 && i <= N - 2 && j >= 1 && j <= 2)
    #pragma unroll
    for (int c = 0; c < 4; ++c) s1 += d_FUV[c] * Hyb[(size_t)(i - 1) * SY + (j - 1 + c)];
  if (i >= 2 && i <= N - 2 && j >= N - 2 && j <= N - 1)
    #pragma unroll
    for (int c = 0; c < 4; ++c) s1 += d_FDV[c] * Hyb[(size_t)i * SY + (j - 4 + c)];

  const size_t idx = (size_t)b * (N + 1) * (N + 1) + (size_t)i * (N + 1) + j;
  Eo[idx] = E[idx] + CFL * (s1 - s2);
}

// ---------------------------------------------------------------------------
// Faraday: per-point VALU (BW-bound; E is L2-resident after the E-update)
// ---------------------------------------------------------------------------
__global__ void faraday_hx(const float* __restrict__ E, const float* __restrict__ Hx,
                           float* __restrict__ Hxo, const float* __restrict__ pb,
                           const float* __restrict__ pd, const float* __restrict__ pg) {
  const int j = blockIdx.x * blockDim.x + threadIdx.x;
  const int i = blockIdx.y;
  const int b = blockIdx.z;
  if (j > N - 1 || i > N - 2) return;
  float W[3][4];
  makeW(pb[0], pd[0], pg[0], W);
  const float* Eb = E + (size_t)b * (N + 1) * (N + 1);
  // pull the deepest E row of this stencil into cache early (global_prefetch_b8)
  __builtin_prefetch(&Eb[(size_t)(i + 4 <= N ? i + 4 : N) * (N + 1) + j], 0, 3);
  float s3 = 0.f;
  if (j >= 1 && j <= N - 2) {
    #pragma unroll
    for (int r = 0; r < 3; ++r)
      #pragma unroll
      for (int c = 0; c < 4; ++c)
        s3 += W[r][c] * Eb[(size_t)(i + r) * (N + 1) + (j - 1 + c)];
  }
  if (i <= N - 4)
    #pragma unroll
    for (int r = 0; r < 4; ++r) s3 += d_KEF[r] * Eb[(size_t)(i + 1 + r) * (N + 1) + j];
  if (i >= 2)
    #pragma unroll
    for (int r = 0; r < 4; ++r) s3 += d_KEB[r] * Eb[(size_t)(i - 2 + r) * (N + 1) + (j + 1)];
  const size_t idx = (size_t)b * (N - 1) * N + (size_t)i * N + j;
  Hxo[idx] = Hx[idx] - CFL * s3;
}

__global__ void faraday_hy(const float* __restrict__ E, const float* __restrict__ Hy,
                           float* __restrict__ Hyo, const float* __restrict__ pb,
                           const float* __restrict__ pd, const float* __restrict__ pg) {
  const int j = blockIdx.x * blockDim.x + threadIdx.x;
  const int i = blockIdx.y;
  const int b = blockIdx.z;
  if (j > N - 2 || i > N - 1) return;
  float W[3][4];
  makeW(pb[0], pd[0], pg[0], W);
  const float* Eb = E + (size_t)b * (N + 1) * (N + 1);
  __builtin_prefetch(&Eb[(size_t)(i + 1 <= N ? i + 1 : N) * (N + 1) + j], 0, 3);
  float s4 = 0.f;
  if (i >= 1 && i <= N - 2) {
    #pragma unroll
    for (int r = 0; r < 4; ++r)
      #pragma unroll
      for (int c = 0; c < 3; ++c)
        s4 += W[c][r] * Eb[(size_t)(i - 1 + r) * (N + 1) + (j + c)];
  }
  if (j <= N - 4)
    #pragma unroll
    for (int c = 0; c < 4; ++c) s4 += d_KEF[c] * Eb[(size_t)i * (N + 1) + (j + 1 + c)];
  if (j >= 2)
    #pragma unroll
    for (int c = 0; c < 4; ++c) s4 += d_KEB[c] * Eb[(size_t)(i + 1) * (N + 1) + (j - 2 + c)];
  const size_t idx = (size_t)b * N * (N - 1) + (size_t)i * (N - 1) + j;
  Hyo[idx] = Hy[idx] + CFL * s4;
}

// ---------------------------------------------------------------------------
extern "C" void kernel_launch(void* const* d_in, const int* in_sizes, int n_in,
                              void* d_out, int out_size, void* d_ws, size_t ws_size,
                              hipStream_t stream) {
  (void)in_sizes; (void)n_in; (void)out_size; (void)d_ws; (void)ws_size;
  const float* E  = (const float*)d_in[0];
  const float* Hx = (const float*)d_in[1];
  const float* Hy = (const float*)d_in[2];
  const float* pb = (const float*)d_in[3];
  const float* pd = (const float*)d_in[4];
  const float* pg = (const float*)d_in[5];

  const size_t szE  = (size_t)BATCH * (N + 1) * (N + 1);
  const size_t szHx = (size_t)BATCH * (N - 1) * N;
  const size_t szHy = (size_t)BATCH * N * (N - 1);
  float* out = (float*)d_out;
  float* E2 = out;          float* Hx2 = E2 + szE;  float* Hy2 = Hx2 + szHx;
  float* E3 = Hy2 + szHy;   float* Hx3 = E3 + szE;  float* Hy3 = Hx3 + szHx;
  float* E4 = Hy3 + szHy;   float* Hx4 = E4 + szE;  float* Hy4 = Hx4 + szHx;

  const dim3 tGrid(NTILE, NTILE, BATCH), tBlk(32, 1, 1);
  const dim3 eBlk(64, 4, 1), eGrid((N + 64) / 64, (N + 4) / 4, BATCH);
  const dim3 fBlk(256, 1, 1);
  const dim3 fxGrid((N + 255) / 256, N - 1, BATCH);
  const dim3 fyGrid((N - 1 + 255) / 256, N, BATCH);

  auto amper = [&](const float* Ein, const float* hx, const float* hy, float* Eo) {
    amper_tile<<<tGrid, tBlk, 0, stream>>>(Ein, hx, hy, Eo, pb, pd, pg);
    amper_edge<<<eGrid, eBlk, 0, stream>>>(Ein, hx, hy, Eo, pb, pd, pg);
  };

  amper(E, Hx, Hy, E2);
  faraday_hx<<<fxGrid, fBlk, 0, stream>>>(E2, Hx, Hx2, pb, pd, pg);
  faraday_hy<<<fyGrid, fBlk, 0, stream>>>(E2, Hy, Hy2, pb, pd, pg);

  amper(E2, Hx2, Hy2, E3);
  faraday_hx<<<fxGrid, fBlk, 0, stream>>>(E3, Hx2, Hx3, pb, pd, pg);
  faraday_hy<<<fyGrid, fBlk, 0, stream>>>(E3, Hy2, Hy3, pb, pd, pg);

  amper(E3, Hx3, Hy3, E4);
  faraday_hx<<<fxGrid, fBlk, 0, stream>>>(E4, Hx3, Hx4, pb, pd, pg);
  faraday_hy<<<fyGrid, fBlk, 0, stream>>>(E4, Hy3, Hy4, pb, pd, pg);
}